// SRLFeatureExtractor_89300960018792
// MI455X (gfx1250) — compile-verified
//
#include <hip/hip_runtime.h>

// SRL feature extractor: S=512, L=256, H=768, P=8, K=8
// e[role][s][h] = (1/n_pred) * sum_p [pred & cnt_p>0] * (1/cnt_p) * sum_k valid_pk * emb[s, idx_pk, h]
//
// Block-diagonal f32 GEMM per (s, role):
//   W(16x64, 0/1 weights, row = predicate p) x G(64x768, gathered rows) = span_sum(16x768)
// via V_WMMA_F32_16X16X4_F32 (full-precision f32 WMMA), 48 N-tiles x 16 K-chunks.
// Memory-bound (~200-300 MB gathers -> ~10 us floor @ 23.3 TB/s); this version
// software-pipelines each tile: issue all 32 B-loads, then the 16-WMMA chain,
// so up to 32 loads are in flight per wave instead of 2.

typedef float v2f __attribute__((ext_vector_type(2)));
typedef float v8f __attribute__((ext_vector_type(8)));

#define S_ 512
#define L_ 256
#define H_ 768

__global__ __launch_bounds__(256) void srl_pool_wmma(
    const float* __restrict__ emb,
    const int* __restrict__ idxV,  const unsigned char* __restrict__ mV,
    const int* __restrict__ idxA0, const unsigned char* __restrict__ mA0,
    const int* __restrict__ idxA1, const unsigned char* __restrict__ mA1,
    const unsigned char* __restrict__ pred,
    float* __restrict__ out)
{
    __shared__ int   s_rowoff[64];   // clamped idx * H (element offset of gathered row)
    __shared__ float s_w[64];        // valid ? 1 : 0, flattened j = p*8 + k
    __shared__ float s_cntmax[8];    // max(cnt_p, 1)
    __shared__ float s_active[8];    // (pred_p && cnt_p>0) ? 1 : 0
    __shared__ float s_npredmax;     // max(n_pred, 1)
    __shared__ float s_prednz;       // (n_pred > 0) ? 1 : 0

    const int s    = blockIdx.x;
    const int role = blockIdx.y;
    const int* idx; const unsigned char* msk;
    if (role == 0)      { idx = idxV;  msk = mV;  }
    else if (role == 1) { idx = idxA0; msk = mA0; }
    else                { idx = idxA1; msk = mA1; }

    const int tid = threadIdx.x;
    if (tid < 64) {
        int iv = idx[s * 64 + tid];
        bool valid = (msk[s * 64 + tid] != 0) && (iv < L_);
        int ic = iv < 0 ? 0 : (iv > (L_ - 1) ? (L_ - 1) : iv);
        s_rowoff[tid] = ic * H_;
        s_w[tid]      = valid ? 1.0f : 0.0f;
    }
    __syncthreads();
    if (tid < 8) {
        float c = 0.0f;
        for (int k = 0; k < 8; ++k) c += s_w[tid * 8 + k];
        s_cntmax[tid] = c > 1.0f ? c : 1.0f;
        s_active[tid] = ((pred[s * 8 + tid] != 0) && (c > 0.0f)) ? 1.0f : 0.0f;
    }
    if (tid == 0) {
        float np = 0.0f;
        for (int p = 0; p < 8; ++p) np += (pred[s * 8 + p] != 0) ? 1.0f : 0.0f;
        s_npredmax = np > 1.0f ? np : 1.0f;
        s_prednz   = np > 0.0f ? 1.0f : 0.0f;
    }
    __syncthreads();

    const int lane  = tid & 31;
    const int wave  = tid >> 5;
    const int mlane = lane & 15;          // M (for A) / N (for B,D) within tile
    const int khalf = (lane >> 4) << 1;   // lanes 0-15: K0/K1; lanes 16-31: K2/K3

    const float* embS = emb + (long)s * (L_ * H_);
    float*       outS = out + (long)role * (S_ * H_) + (long)s * H_;

    // Tile-invariant per-lane state: A-matrix weight pairs and row offsets.
    v2f Aw[16];
    int off0[16], off1[16];
    #pragma unroll
    for (int kc = 0; kc < 16; ++kc) {
        const int j0  = kc * 4;
        const int pch = j0 >> 3;                // predicate owning this chunk
        const int jA  = j0 + khalf;
        const float keep = (mlane == pch) ? 1.0f : 0.0f;
        Aw[kc].x = keep * s_w[jA];
        Aw[kc].y = keep * s_w[jA + 1];
        off0[kc] = s_rowoff[jA];
        off1[kc] = s_rowoff[jA + 1];
    }

    // 48 N-tiles of 16 columns; 8 waves, 6 tiles each.
    for (int t = wave; t < H_ / 16; t += 8) {
        const int col = t * 16 + mlane;

        // Phase 1: issue all 32 gather loads for this tile (max MLP).
        float b0[16], b1[16];
        #pragma unroll
        for (int kc = 0; kc < 16; ++kc) {
            b0[kc] = embS[off0[kc] + col];
            b1[kc] = embS[off1[kc] + col];
        }

        // Phase 2: 16-deep WMMA accumulation chain.
        v8f acc = {};
        #pragma unroll
        for (int kc = 0; kc < 16; ++kc) {
            v2f B;
            B.x = b0[kc];
            B.y = b1[kc];
            acc = __builtin_amdgcn_wmma_f32_16x16x4_f32(
                      false, Aw[kc], false, B, (short)0, acc, false, false);
        }

        // D: VGPR r, lanes 0-15 = span_sum[p=r, n=lane]; rows 8-15 are zero.
        if (lane < 16) {
            float sum = 0.0f;
            #pragma unroll
            for (int p = 0; p < 8; ++p)
                sum += s_active[p] * (acc[p] / s_cntmax[p]);
            outS[t * 16 + mlane] = s_prednz * (sum / s_npredmax);
        }
    }
}

extern "C" void kernel_launch(void* const* d_in, const int* in_sizes, int n_in,
                              void* d_out, int out_size, void* d_ws, size_t ws_size,
                              hipStream_t stream) {
    const float*         emb   = (const float*)d_in[0];
    const int*           idxV  = (const int*)d_in[1];
    const unsigned char* mV    = (const unsigned char*)d_in[2];
    const int*           idxA0 = (const int*)d_in[3];
    const unsigned char* mA0   = (const unsigned char*)d_in[4];
    const int*           idxA1 = (const int*)d_in[5];
    const unsigned char* mA1   = (const unsigned char*)d_in[6];
    const unsigned char* pred  = (const unsigned char*)d_in[7];
    float*               out   = (float*)d_out;

    dim3 grid(S_, 3);
    srl_pool_wmma<<<grid, 256, 0, stream>>>(emb, idxV, mV, idxA0, mA0, idxA1, mA1, pred, out);
}